// VolRender_69879117906386
// MI455X (gfx1250) — compile-verified
//
#include <hip/hip_runtime.h>

// ---------- types ----------
typedef __attribute__((ext_vector_type(2))) float v2f;
typedef __attribute__((ext_vector_type(8))) float v8f;

#define DVOX   64
#define VOXN   (DVOX*DVOX*DVOX)       // 262144
#define RESR   128
#define RESI   256
#define NPTS   64
#define CFEAT  16

// =====================================================================
// Kernel 1: ray generation + trilinear sampling + emission/absorption
// one thread per render pixel (b,h,w); loops 64 depth points.
// outputs img (B,16,128,128) and opacity (B,128,128) into workspace.
// =====================================================================
__global__ __launch_bounds__(256) void render_kernel(
    const float* __restrict__ Kc, const float* __restrict__ R,
    const float* __restrict__ T,  const float* __restrict__ feat,
    const float* __restrict__ dens, float* __restrict__ img,
    float* __restrict__ opac)
{
    int idx = blockIdx.x * blockDim.x + threadIdx.x;   // 0..32767
    int b   = idx >> 14;
    int pix = idx & 16383;
    int h   = pix >> 7;
    int w   = pix & 127;

    const float* Kb = Kc + b * 9;
    const float* Rb = R  + b * 9;
    const float* Tb = T  + b * 3;

    // K/DOWN, scale=RES_RENDER/2=64
    float fx = Kb[0] * 0.5f / 64.f;
    float fy = Kb[4] * 0.5f / 64.f;
    float px = -(Kb[2] * 0.5f - 64.f) / 64.f;
    float py = -(Kb[5] * 0.5f - 64.f) / 64.f;

    float xs = (127.f - 2.f * (float)w) / 128.f;
    float ys = (127.f - 2.f * (float)h) / 128.f;
    float ax = (xs - px) / fx;   // Xc = ax * z
    float ay = (ys - py) / fy;   // Yc = ay * z
    float T0 = Tb[0], T1 = Tb[1], T2 = Tb[2];

    const float inv_loc = 64.f / 63.f;   // 1/loc_scale
    float acc[CFEAT];
#pragma unroll
    for (int c = 0; c < CFEAT; ++c) acc[c] = 0.f;
    float absorb = 1.f, om = 1.f;

    const float* db = dens + b * VOXN;
    const float* fb = feat + b * CFEAT * VOXN;

    for (int p = 0; p < NPTS; ++p) {
        float z  = 1.f + 2.f * (float)p / 63.f;
        float d0 = ax * z + T0;
        float d1 = ay * z + T1;
        float d2 = z - T2;
        // world = Rp @ (cam - Tp);  Rp[j,i] = s_i * R[i,j], s = (-1,-1,1)
        float wx = -d0 * Rb[0] - d1 * Rb[3] + d2 * Rb[6];
        float wy = -d0 * Rb[1] - d1 * Rb[4] + d2 * Rb[7];
        float wz = -d0 * Rb[2] - d1 * Rb[5] + d2 * Rb[8];

        float ix = (wx * inv_loc + 1.f) * 31.5f;
        float iy = (wy * inv_loc + 1.f) * 31.5f;
        float iz = (wz * inv_loc + 1.f) * 31.5f;
        float x0f = floorf(ix), y0f = floorf(iy), z0f = floorf(iz);
        float tx = ix - x0f, ty = iy - y0f, tz = iz - z0f;
        int x0 = (int)x0f, y0 = (int)y0f, z0 = (int)z0f;

        int   cidx[8];
        float cw[8];
        int q = 0;
#pragma unroll
        for (int dz = 0; dz < 2; ++dz)
#pragma unroll
        for (int dy = 0; dy < 2; ++dy)
#pragma unroll
        for (int dx = 0; dx < 2; ++dx) {
            int xi = x0 + dx, yi = y0 + dy, zi = z0 + dz;
            bool valid = (xi >= 0) & (xi < DVOX) & (yi >= 0) & (yi < DVOX) &
                         (zi >= 0) & (zi < DVOX);
            float wgt = (dx ? tx : 1.f - tx) * (dy ? ty : 1.f - ty) *
                        (dz ? tz : 1.f - tz);
            int xc = min(max(xi, 0), DVOX - 1);
            int yc = min(max(yi, 0), DVOX - 1);
            int zc = min(max(zi, 0), DVOX - 1);
            cidx[q] = (zc * DVOX + yc) * DVOX + xc;
            cw[q]   = valid ? wgt : 0.f;
            ++q;
        }

        float sigma = 0.f;
#pragma unroll
        for (int k = 0; k < 8; ++k) sigma += db[cidx[k]] * cw[k];

        float wgtp = sigma * absorb;
#pragma unroll
        for (int c = 0; c < CFEAT; ++c) {
            const float* fc = fb + c * VOXN;
            float v = 0.f;
#pragma unroll
            for (int k = 0; k < 8; ++k) v += fc[cidx[k]] * cw[k];
            acc[c] += v * wgtp;
        }
        absorb *= (1.f + 1e-10f - sigma);
        om     *= (1.f - sigma);
    }

#pragma unroll
    for (int c = 0; c < CFEAT; ++c)
        img[((b * CFEAT + c) << 14) + pix] = acc[c];
    opac[(b << 14) + pix] = 1.f - om;
}

// =====================================================================
// Kernel 2: bilinear 2x upsample of opacity -> mask (half-pixel centers,
// clamp-to-edge == jax.image.resize 'bilinear' after renormalization)
// =====================================================================
__global__ __launch_bounds__(256) void mask_kernel(
    const float* __restrict__ opac, float* __restrict__ out)
{
    int i = blockIdx.x * blockDim.x + threadIdx.x;   // B*256*256
    if (i >= 2 * RESI * RESI) return;
    int b = i >> 16;
    int r = i & 65535;
    int y = r >> 8;
    int x = r & 255;
    float sfy = 0.5f * (float)y - 0.25f;
    float sfx = 0.5f * (float)x - 0.25f;
    float fy0 = floorf(sfy), fx0 = floorf(sfx);
    float ty = sfy - fy0, tx = sfx - fx0;
    int y0 = (int)fy0, x0 = (int)fx0;
    int y0c = min(max(y0, 0), 127), y1c = min(max(y0 + 1, 0), 127);
    int x0c = min(max(x0, 0), 127), x1c = min(max(x0 + 1, 0), 127);
    const float* ob = opac + (b << 14);
    float v00 = ob[y0c * 128 + x0c], v01 = ob[y0c * 128 + x1c];
    float v10 = ob[y1c * 128 + x0c], v11 = ob[y1c * 128 + x1c];
    out[i] = (1.f - ty) * ((1.f - tx) * v00 + tx * v01) +
             ty * ((1.f - tx) * v10 + tx * v11);
}

// =====================================================================
// Kernel 3: 4x4 stride-2 transposed conv 16->16 via V_WMMA_F32_16X16X4_F32
// one wave per tile of 16 same-parity output columns of one row.
// D[m][n] = sum_{i,dy,dx} w_up[i][m][ky0-2dy][kx0-2dx] * img[i][iy0+dy][ix0(n)+dx]
// K = 16 chunks of 4 (chunk = input channel i; within chunk k=(dy,dx)).
// Output x1 = deconv + b_up  (pre-BN).
// =====================================================================
__global__ __launch_bounds__(128) void deconv_wmma_kernel(
    const float* __restrict__ img, const float* __restrict__ wup,
    const float* __restrict__ bup, float* __restrict__ x1)
{
    int wave = blockIdx.x * 4 + (threadIdx.x >> 5);
    int lane = threadIdx.x & 31;
    int n  = lane & 15;     // A-row m / B-col n (same lane field)
    int hi = lane >> 4;     // selects K pair {2hi, 2hi+1} => dy = hi

    int b   = wave >> 12;        // 4096 tiles per batch
    int rem = wave & 4095;
    int y   = rem >> 4;          // output row 0..255
    int tt  = rem & 15;
    int rx  = tt >> 3;           // x parity
    int xb  = tt & 7;
    int x   = rx + (xb << 5) + (n << 1);   // this lane's output column

    int iy0 = ((y + 1) >> 1) - 1;
    int ky0 = y + 1 - 2 * iy0;             // 3 (even y) or 2 (odd y)
    int iyv = iy0 + hi;
    int kyv = ky0 - 2 * hi;
    int ix0 = ((x + 1) >> 1) - 1;
    int kx0 = 3 - rx;                      // dx=0 tap
    int kx1 = kx0 - 2;                     // dx=1 tap

    bool vy  = (iyv >= 0) & (iyv < RESR);
    int  iyc = min(max(iyv, 0), RESR - 1);
    bool vx0 = (ix0 >= 0) & (ix0 < RESR);
    bool vx1 = (ix0 + 1 >= 0) & (ix0 + 1 < RESR);
    int  ix0c = min(max(ix0, 0), RESR - 1);
    int  ix1c = min(max(ix0 + 1, 0), RESR - 1);

    v8f c = {};
#pragma unroll 4
    for (int i = 0; i < CFEAT; ++i) {
        v2f a, bm;
        int wb = ((i * 16 + n) * 4 + kyv) * 4;     // w_up[i][m][kyv][.]
        a.x = wup[wb + kx0];
        a.y = wup[wb + kx1];
        const float* ip = img + ((b * CFEAT + i) << 14) + (iyc << 7);
        bm.x = (vy && vx0) ? ip[ix0c] : 0.f;
        bm.y = (vy && vx1) ? ip[ix1c] : 0.f;
        c = __builtin_amdgcn_wmma_f32_16x16x4_f32(false, a, false, bm,
                                                  (short)0, c, false, false);
    }
#pragma unroll
    for (int v = 0; v < 8; ++v) {
        int m = v + (hi << 3);
        x1[((b * CFEAT + m) << 16) + (y << 8) + x] = c[v] + bup[m];
    }
}

// =====================================================================
// Kernel 4/5: batch-norm stats (one block per channel) + apply with lrelu
// =====================================================================
__global__ __launch_bounds__(256) void bn_stats_kernel(
    const float* __restrict__ x, float* __restrict__ mean,
    float* __restrict__ istd, int C, int HW)
{
    int ch = blockIdx.x;
    int N  = 2 * HW;   // B = 2
    float s = 0.f, sq = 0.f;
    for (int i = threadIdx.x; i < N; i += blockDim.x) {
        int bb = (i >= HW) ? 1 : 0;
        int r  = i - bb * HW;
        float v = x[(bb * C + ch) * HW + r];
        s += v; sq += v * v;
    }
    __shared__ float sh1[256], sh2[256];
    sh1[threadIdx.x] = s; sh2[threadIdx.x] = sq;
    __syncthreads();
    for (int off = 128; off > 0; off >>= 1) {
        if ((int)threadIdx.x < off) {
            sh1[threadIdx.x] += sh1[threadIdx.x + off];
            sh2[threadIdx.x] += sh2[threadIdx.x + off];
        }
        __syncthreads();
    }
    if (threadIdx.x == 0) {
        float m = sh1[0] / (float)N;
        float v = sh2[0] / (float)N - m * m;
        mean[ch] = m;
        istd[ch] = rsqrtf(v + 1e-5f);
    }
}

__global__ __launch_bounds__(256) void bn_apply_kernel(
    float* __restrict__ x, const float* __restrict__ mean,
    const float* __restrict__ istd, const float* __restrict__ g,
    const float* __restrict__ beta, int C, int HW, int total)
{
    int i = blockIdx.x * blockDim.x + threadIdx.x;
    if (i >= total) return;
    int ch = (i / HW) % C;
    float v = (x[i] - mean[ch]) * istd[ch] * g[ch] + beta[ch];
    x[i] = (v >= 0.f) ? v : 0.01f * v;
}

// =====================================================================
// Kernel 6/7: 3x3 conv (pad 1) as implicit GEMM with V_WMMA_F32_16X16X4_F32
// M = 16 (COUT rows used), N = 16 consecutive pixels, K = CIN*9.
// =====================================================================
template <int CIN, int COUT, bool RELU>
__global__ __launch_bounds__(128) void conv3x3_wmma_kernel(
    const float* __restrict__ xin, const float* __restrict__ wgt,
    const float* __restrict__ bias, float* __restrict__ out)
{
    int wave = blockIdx.x * 4 + (threadIdx.x >> 5);
    int lane = threadIdx.x & 31;
    int n  = lane & 15;      // A-row m / B-col n
    int hi = lane >> 4;

    int b   = wave >> 12;
    int rem = wave & 4095;
    int y   = rem >> 4;
    int x   = ((rem & 15) << 4) + n;

    v8f c = {};
    const int NCH = (CIN * 9) / 4;
#pragma unroll 1
    for (int kc = 0; kc < NCH; ++kc) {
        int k0 = kc * 4 + 2 * hi;
        v2f a, bm;
        {
            int k = k0;
            int i = k / 9; int r = k - i * 9; int ky = r / 3; int kx = r - ky * 3;
            a.x = (n < COUT) ? wgt[((n * CIN + i) * 3 + ky) * 3 + kx] : 0.f;
            int yy = y + ky - 1, xx = x + kx - 1;
            bool v = (yy >= 0) & (yy < RESI) & (xx >= 0) & (xx < RESI);
            int yyc = min(max(yy, 0), RESI - 1), xxc = min(max(xx, 0), RESI - 1);
            bm.x = v ? xin[((b * CIN + i) << 16) + (yyc << 8) + xxc] : 0.f;
        }
        {
            int k = k0 + 1;
            int i = k / 9; int r = k - i * 9; int ky = r / 3; int kx = r - ky * 3;
            a.y = (n < COUT) ? wgt[((n * CIN + i) * 3 + ky) * 3 + kx] : 0.f;
            int yy = y + ky - 1, xx = x + kx - 1;
            bool v = (yy >= 0) & (yy < RESI) & (xx >= 0) & (xx < RESI);
            int yyc = min(max(yy, 0), RESI - 1), xxc = min(max(xx, 0), RESI - 1);
            bm.y = v ? xin[((b * CIN + i) << 16) + (yyc << 8) + xxc] : 0.f;
        }
        c = __builtin_amdgcn_wmma_f32_16x16x4_f32(false, a, false, bm,
                                                  (short)0, c, false, false);
    }
#pragma unroll
    for (int v = 0; v < 8; ++v) {
        int m = v + (hi << 3);
        if (m < COUT) {
            float val = c[v] + bias[m];
            if (RELU) val = fmaxf(val, 0.f);
            out[((b * COUT + m) << 16) + (y << 8) + x] = val;
        }
    }
}

// =====================================================================
// launcher
// =====================================================================
extern "C" void kernel_launch(void* const* d_in, const int* in_sizes, int n_in,
                              void* d_out, int out_size, void* d_ws, size_t ws_size,
                              hipStream_t stream)
{
    (void)in_sizes; (void)n_in; (void)out_size; (void)ws_size;
    const float* Kc    = (const float*)d_in[0];
    const float* R     = (const float*)d_in[1];
    const float* T     = (const float*)d_in[2];
    const float* feat  = (const float*)d_in[3];
    const float* dens  = (const float*)d_in[4];
    const float* w_up  = (const float*)d_in[5];
    const float* b_up  = (const float*)d_in[6];
    const float* bn1_g = (const float*)d_in[7];
    const float* bn1_b = (const float*)d_in[8];
    const float* w_c1  = (const float*)d_in[9];
    const float* b_c1  = (const float*)d_in[10];
    const float* bn2_g = (const float*)d_in[11];
    const float* bn2_b = (const float*)d_in[12];
    const float* w_c2  = (const float*)d_in[13];
    const float* b_c2  = (const float*)d_in[14];

    float* ws    = (float*)d_ws;
    float* img   = ws;                       // B*16*128*128 = 524288
    float* opac  = img  + 524288;            // B*128*128    = 32768
    float* x1    = opac + 32768;             // B*16*256*256 = 2097152
    float* x2    = x1   + 2097152;           // B*8*256*256  = 1048576
    float* stats = x2   + 1048576;           // 48 floats
    float* rgb   = (float*)d_out;            // B*3*256*256
    float* mask  = rgb + 2 * 3 * 65536;      // B*1*256*256

    // 1) render
    render_kernel<<<128, 256, 0, stream>>>(Kc, R, T, feat, dens, img, opac);
    // 2) mask resize
    mask_kernel<<<512, 256, 0, stream>>>(opac, mask);
    // 3) deconv (WMMA f32) -> x1 (+bias)
    deconv_wmma_kernel<<<2048, 128, 0, stream>>>(img, w_up, b_up, x1);
    // 4) BN1 stats + apply (lrelu) in place
    bn_stats_kernel<<<16, 256, 0, stream>>>(x1, stats, stats + 16, 16, 65536);
    bn_apply_kernel<<<8192, 256, 0, stream>>>(x1, stats, stats + 16, bn1_g, bn1_b,
                                              16, 65536, 2097152);
    // 5) conv1 (WMMA f32) -> x2 (+bias)
    conv3x3_wmma_kernel<16, 8, false><<<2048, 128, 0, stream>>>(x1, w_c1, b_c1, x2);
    // 6) BN2 stats + apply (lrelu) in place
    bn_stats_kernel<<<8, 256, 0, stream>>>(x2, stats + 32, stats + 40, 8, 65536);
    bn_apply_kernel<<<4096, 256, 0, stream>>>(x2, stats + 32, stats + 40, bn2_g, bn2_b,
                                              8, 65536, 1048576);
    // 7) conv2 (WMMA f32) + ReLU -> rgb
    conv3x3_wmma_kernel<8, 3, true><<<2048, 128, 0, stream>>>(x2, w_c2, b_c2, rgb);
}